// GCN_18554258719193
// MI455X (gfx1250) — compile-verified
//
#include <hip/hip_runtime.h>
#include <math.h>

// GCN 2-layer forward for MI455X (gfx1250, wave32).
// GEMMs use V_WMMA_F32_16X16X4_F32 (full f32 precision, D = A*B + C).
// Propagates: one wave32 per edge, float4/float2 gathers + f32 global atomics.
//
// Workspace layout (floats), requires ws_size >= (324*N + E_knn)*4 bytes ~ 66.4 MB:
//   deg[N] | dis[N] | degw[N] | disw[N] | hlin[N*128] | h[N*128] | h2[N*64] | wknn[E_knn]

typedef float v2f __attribute__((ext_vector_type(2)));
typedef float v8f __attribute__((ext_vector_type(8)));

#define TPB 256

// ---------------- small elementwise / degree kernels ----------------

__global__ __launch_bounds__(TPB) void k_init_deg(float* deg, float* degw, int n) {
    int i = blockIdx.x * TPB + threadIdx.x;
    if (i < n) { deg[i] = 1.0f; degw[i] = 1.0f; }   // self-loop contributes 1
}

__global__ __launch_bounds__(TPB) void k_count_edges(const int* __restrict__ idx,
                                                     float* __restrict__ deg, int e) {
    int i = blockIdx.x * TPB + threadIdx.x;
    if (i < e) atomicAdd(&deg[idx[i]], 1.0f);
}

__global__ __launch_bounds__(TPB) void k_rsqrt(const float* __restrict__ a,
                                               float* __restrict__ o, int n) {
    int i = blockIdx.x * TPB + threadIdx.x;
    if (i < n) o[i] = rsqrtf(a[i]);
}

__global__ __launch_bounds__(TPB) void k_relu(float* h, long long n) {
    long long i = (long long)blockIdx.x * TPB + threadIdx.x;
    if (i < n) h[i] = fmaxf(h[i], 0.0f);
}

// out[i,:] = s[i]^2 * in[i,:]   (self-loop term; also fully initializes out)
// D4 = feature dim / 4
__global__ __launch_bounds__(TPB) void k_selfloop_scale(const float* __restrict__ s,
                                                        const float* __restrict__ hin,
                                                        float* __restrict__ hout,
                                                        int n, int d4) {
    long long t = (long long)blockIdx.x * TPB + threadIdx.x;
    if (t >= (long long)n * d4) return;
    int i = (int)(t / d4);
    float sv = s[i]; sv = sv * sv;
    float4 v = ((const float4*)hin)[t];
    float4 r; r.x = sv * v.x; r.y = sv * v.y; r.z = sv * v.z; r.w = sv * v.w;
    ((float4*)hout)[t] = r;
}

// ---------------- WMMA GEMM: out[M,Nc] = A[M,K] @ W[Nc,K]^T + bias ----------------
// One wave per 16x16 output tile. M, Nc multiples of 16; K multiple of 4.
// A 16x4 f32 layout: v0 = K=kb+2*half, v1 = K=kb+2*half+1, row = m0 + (lane&15)
// B 4x16 f32 layout: same K striping, col = n0 + (lane&15), B[k][n] = W[n*K + k]
// D layout: vgpr r -> row m0 + r + 8*half, col n0 + (lane&15)
__global__ __launch_bounds__(TPB) void k_gemm_bias_wmma(const float* __restrict__ A,
                                                        const float* __restrict__ W,
                                                        const float* __restrict__ bias,
                                                        float* __restrict__ out,
                                                        int M, int K, int Nc) {
    int wid = (blockIdx.x * TPB + threadIdx.x) >> 5;    // wave-uniform
    int ntile = Nc >> 4;
    int total = (M >> 4) * ntile;
    if (wid >= total) return;                           // uniform exit: EXEC stays all-1s

    int mt = wid / ntile, nt = wid - mt * ntile;
    int m0 = mt << 4, n0 = nt << 4;
    int lane = threadIdx.x & 31;
    int half = lane >> 4;        // 0: K pair {kb,kb+1}; 1: {kb+2,kb+3}
    int l16  = lane & 15;

    const float* arow = A + (long long)(m0 + l16) * K;
    const float* wrow = W + (long long)(n0 + l16) * K;

    v8f c = {0.f, 0.f, 0.f, 0.f, 0.f, 0.f, 0.f, 0.f};
    for (int kb = 0; kb < K; kb += 4) {
        int k = kb + 2 * half;                   // even -> 8B aligned
        v2f a = *(const v2f*)(arow + k);
        v2f b = *(const v2f*)(wrow + k);
        c = __builtin_amdgcn_wmma_f32_16x16x4_f32(false, a, false, b,
                                                  (short)0, c, false, false);
    }

    float bv = bias[n0 + l16];
    long long colbase = (long long)n0 + l16;
#pragma unroll
    for (int r = 0; r < 8; ++r) {
        int m = m0 + r + 8 * half;
        out[(long long)m * Nc + colbase] = c[r] + bv;
    }
}

// ---------------- propagate layer 1: 128 features, wave per edge ----------------
__global__ __launch_bounds__(TPB) void k_prop1_edges(const int* __restrict__ row,
                                                     const int* __restrict__ col,
                                                     const float* __restrict__ dis,
                                                     const float* __restrict__ hlin,
                                                     float* __restrict__ hagg, int E) {
    long long t = (long long)blockIdx.x * TPB + threadIdx.x;
    int e = (int)(t >> 5);
    if (e >= E) return;                                 // wave-uniform
    int lane = (int)(t & 31);
    int r = row[e], c = col[e];
    float nrm = dis[r] * dis[c];
    float4 v = ((const float4*)(hlin + (long long)r * 128))[lane];
    float* dst = hagg + (long long)c * 128 + lane * 4;
    atomicAdd(dst + 0, nrm * v.x);
    atomicAdd(dst + 1, nrm * v.y);
    atomicAdd(dst + 2, nrm * v.z);
    atomicAdd(dst + 3, nrm * v.w);
}

// ---------------- knn edge weights + weighted-degree scatter ----------------
__global__ __launch_bounds__(TPB) void k_knn_weights(const int* __restrict__ rk,
                                                     const int* __restrict__ ck,
                                                     const float* __restrict__ h,
                                                     const float* __restrict__ alpha,
                                                     float* __restrict__ wknn,
                                                     float* __restrict__ degw, int E) {
    long long t = (long long)blockIdx.x * TPB + threadIdx.x;
    int e = (int)(t >> 5);
    if (e >= E) return;
    int lane = (int)(t & 31);
    int a = rk[e], b = ck[e];
    float4 va = ((const float4*)(h + (long long)a * 128))[lane];
    float4 vb = ((const float4*)(h + (long long)b * 128))[lane];
    float dx = va.x - vb.x, dy = va.y - vb.y, dz = va.z - vb.z, dw = va.w - vb.w;
    float ss = dx * dx + dy * dy + dz * dz + dw * dw;
#pragma unroll
    for (int off = 16; off > 0; off >>= 1) ss += __shfl_xor(ss, off, 32);
    if (lane == 0) {
        float wv = 0.0f;
        if (a != b) {
            float nv = fmaxf(sqrtf(ss), 1e-12f);        // clip(norm, 1e-12)
            wv = alpha[0] * sqrtf(nv);                  // ^(P-2) with P=2.5
        }
        wknn[e] = wv;
        if (wv != 0.0f) atomicAdd(&degw[b], wv);
    }
}

// ---------------- propagate layer 2: 64 features, wave per edge ----------------
// wknn == nullptr -> edge weight 1 (org + self-loop path handled separately)
__global__ __launch_bounds__(TPB) void k_prop2_edges(const int* __restrict__ row,
                                                     const int* __restrict__ col,
                                                     const float* __restrict__ disw,
                                                     const float* __restrict__ wknn,
                                                     const float* __restrict__ h2,
                                                     float* __restrict__ out, int E) {
    long long t = (long long)blockIdx.x * TPB + threadIdx.x;
    int e = (int)(t >> 5);
    if (e >= E) return;
    int lane = (int)(t & 31);
    float wv = wknn ? wknn[e] : 1.0f;                   // wave-uniform
    if (wv == 0.0f) return;                             // masked self-edge
    int r = row[e], c = col[e];
    float nrm = disw[r] * disw[c] * wv;
    float2 v = ((const float2*)(h2 + (long long)r * 64))[lane];
    float* dst = out + (long long)c * 64 + lane * 2;
    atomicAdd(dst + 0, nrm * v.x);
    atomicAdd(dst + 1, nrm * v.y);
}

// ---------------- host launcher ----------------

static inline int nblk(long long threads) { return (int)((threads + TPB - 1) / TPB); }

extern "C" void kernel_launch(void* const* d_in, const int* in_sizes, int n_in,
                              void* d_out, int out_size, void* d_ws, size_t ws_size,
                              hipStream_t stream) {
    (void)n_in; (void)out_size; (void)ws_size;
    const float* x      = (const float*)d_in[0];
    const int*   ei_org = (const int*)d_in[1];
    const int*   ei_knn = (const int*)d_in[2];
    const float* alpha  = (const float*)d_in[3];
    const float* W1     = (const float*)d_in[4];
    const float* b1     = (const float*)d_in[5];
    const float* W2     = (const float*)d_in[6];
    const float* b2     = (const float*)d_in[7];
    float* out = (float*)d_out;
    float* ws  = (float*)d_ws;

    const int N = 50000, DIN = 128, DHID = 128, DOUT = 64;
    const int Eo = in_sizes[1] / 2;
    const int Ek = in_sizes[2] / 2;

    float* deg  = ws;
    float* dis  = ws + (long long)N;
    float* degw = ws + 2LL * N;
    float* disw = ws + 3LL * N;
    float* hlin = ws + 4LL * N;
    float* h    = hlin + (long long)N * DHID;
    float* h2   = h + (long long)N * DHID;
    float* wknn = h2 + (long long)N * DOUT;

    const int* row_o = ei_org;           // edge_index_org[0]
    const int* col_o = ei_org + Eo;      // edge_index_org[1]
    const int* row_k = ei_knn;
    const int* col_k = ei_knn + Ek;

    // ---- layer 1 norm: deg from rows (+loop), dis = deg^-1/2 ----
    k_init_deg<<<nblk(N), TPB, 0, stream>>>(deg, degw, N);
    k_count_edges<<<nblk(Eo), TPB, 0, stream>>>(row_o, deg, Eo);
    k_rsqrt<<<nblk(N), TPB, 0, stream>>>(deg, dis, N);

    // ---- hlin = x @ W1^T + b1 (WMMA f32) ----
    long long w1waves = (long long)(N / 16) * (DHID / 16);
    k_gemm_bias_wmma<<<nblk(w1waves * 32), TPB, 0, stream>>>(x, W1, b1, hlin, N, DIN, DHID);

    // ---- propagate 1: self-loop init + edge scatter, then relu ----
    k_selfloop_scale<<<nblk((long long)N * 32), TPB, 0, stream>>>(dis, hlin, h, N, 32);
    k_prop1_edges<<<nblk((long long)Eo * 32), TPB, 0, stream>>>(row_o, col_o, dis, hlin, h, Eo);
    k_relu<<<nblk((long long)N * DHID), TPB, 0, stream>>>(h, (long long)N * DHID);

    // ---- layer 2 weights: degw from cols (org weight 1, knn weight alpha*dist^0.5) ----
    k_count_edges<<<nblk(Eo), TPB, 0, stream>>>(col_o, degw, Eo);
    k_knn_weights<<<nblk((long long)Ek * 32), TPB, 0, stream>>>(row_k, col_k, h, alpha,
                                                                wknn, degw, Ek);
    k_rsqrt<<<nblk(N), TPB, 0, stream>>>(degw, disw, N);

    // ---- h2 = h @ W2^T + b2 (WMMA f32) ----
    long long w2waves = (long long)(N / 16) * (DOUT / 16);
    k_gemm_bias_wmma<<<nblk(w2waves * 32), TPB, 0, stream>>>(h, W2, b2, h2, N, DHID, DOUT);

    // ---- propagate 2: self-loop init + org edges (w=1) + knn edges (w=wknn) ----
    k_selfloop_scale<<<nblk((long long)N * 16), TPB, 0, stream>>>(disw, h2, out, N, 16);
    k_prop2_edges<<<nblk((long long)Eo * 32), TPB, 0, stream>>>(row_o, col_o, disw,
                                                                (const float*)nullptr, h2, out, Eo);
    k_prop2_edges<<<nblk((long long)Ek * 32), TPB, 0, stream>>>(row_k, col_k, disw,
                                                                wknn, h2, out, Ek);
}